// GraphSAGE_20633022890229
// MI455X (gfx1250) — compile-verified
//
#include <hip/hip_runtime.h>
#include <hip/hip_bf16.h>
#include <math.h>

#define NN 50000
#define EE 800000
#define HH 128
#define GG 64
#define LL 3
#define LDSP 132          // LDS row pitch (floats): 132 % 64 = 4 -> conflict-free A reads
#define BN_EPS 1e-5f

typedef __attribute__((ext_vector_type(2))) float v2f;
typedef __attribute__((ext_vector_type(8))) float v8f;

// ---------------------------------------------------------------- utilities
__global__ void zero_f32(float* __restrict__ p, int n) {
  int i = blockIdx.x * blockDim.x + threadIdx.x;
  if (i < n) p[i] = 0.0f;
}

// degree of each destination node (layer-invariant)
__global__ void deg_kernel(const int* __restrict__ ei, float* __restrict__ deg) {
  int e = blockIdx.x * blockDim.x + threadIdx.x;
  if (e < EE) {
    int d = ei[EE + e];
    atomicAdd(&deg[d], 1.0f);
  }
}

// agg[dst, :] += h[src, :]   (one thread per edge x 4-feature chunk; L2-resident)
__global__ void scatter_kernel(const float* __restrict__ hin,
                               const int* __restrict__ ei,
                               float* __restrict__ agg) {
  long long idx = (long long)blockIdx.x * blockDim.x + threadIdx.x;
  if (idx >= (long long)EE * 32) return;
  int e = (int)(idx >> 5);
  int c = ((int)idx & 31) << 2;
  int s = ei[e];
  int d = ei[EE + e];
  const float4 v = *(const float4*)(hin + (long long)s * HH + c);
  float* a = agg + (long long)d * HH + c;
  atomicAdd(a + 0, v.x);
  atomicAdd(a + 1, v.y);
  atomicAdd(a + 2, v.z);
  atomicAdd(a + 3, v.w);
}

// ------------------------------------------------------ fused SAGE layer GEMM
// hout = relu( (agg/deg) @ Wl^T + bl + x @ Wr^T ), plus BN sum / sumsq stats.
// Block = 256 threads = 8 waves; block tile = 16 nodes x 128 features;
// each wave owns one 16x16 tile and issues 64 v_wmma_f32_16x16x4_f32.
__global__ void __launch_bounds__(256) sage_gemm_kernel(
    const float* __restrict__ x, const float* __restrict__ agg,
    const float* __restrict__ deg,
    const float* __restrict__ Wl, const float* __restrict__ blv,
    const float* __restrict__ Wr,
    float* __restrict__ hout,
    float* __restrict__ bnsum, float* __restrict__ bnsq) {
  __shared__ float Aagg[16 * LDSP];
  __shared__ float Ax[16 * LDSP];

  const int t  = threadIdx.x;
  const int m0 = blockIdx.x * 16;

  // stage 16x128 A-tiles (mean-aggregated neighbors, and self features)
#pragma unroll
  for (int i = 0; i < 2; ++i) {
    int lin = t + i * 256;        // float4 index; 512 float4 total per matrix
    int row = lin >> 5;           // 32 float4 per row
    int c4  = (lin & 31) << 2;
    float dscale = 1.0f / fmaxf(deg[m0 + row], 1.0f);
    float4 va = *(const float4*)(agg + (long long)(m0 + row) * HH + c4);
    float4 vx = *(const float4*)(x   + (long long)(m0 + row) * HH + c4);
    float* pa = &Aagg[row * LDSP + c4];
    pa[0] = va.x * dscale; pa[1] = va.y * dscale;
    pa[2] = va.z * dscale; pa[3] = va.w * dscale;
    float* px = &Ax[row * LDSP + c4];
    px[0] = vx.x; px[1] = vx.y; px[2] = vx.z; px[3] = vx.w;
  }
  __syncthreads();

  const int lane = t & 31;
  const int wave = t >> 5;
  const int j0   = wave * 16;
  const int nm   = lane & 15;            // A: M row, B: N column
  const int ksel = (lane >> 4) << 1;     // K pair select per half-wave
  const int j    = j0 + nm;

  const float* arow_a = &Aagg[nm * LDSP];
  const float* arow_x = &Ax[nm * LDSP];
  const float* wlrow  = &Wl[j * HH];
  const float* wrrow  = &Wr[j * HH];
  __builtin_prefetch(wlrow, 0, 0);       // -> global_prefetch_b8
  __builtin_prefetch(wrrow, 0, 0);

  v8f c = {};
#pragma unroll 4
  for (int k0 = 0; k0 < HH; k0 += 4) {
    v2f aa, ax, bb_l, bb_r;
    aa.x = arow_a[k0 + ksel];
    aa.y = arow_a[k0 + ksel + 1];
    ax.x = arow_x[k0 + ksel];
    ax.y = arow_x[k0 + ksel + 1];
    float2 wl2 = *(const float2*)(wlrow + k0 + ksel);
    float2 wr2 = *(const float2*)(wrrow + k0 + ksel);
    bb_l.x = wl2.x; bb_l.y = wl2.y;
    bb_r.x = wr2.x; bb_r.y = wr2.y;
    // D = A(16x4) x B(4x16) + C, fp32
    c = __builtin_amdgcn_wmma_f32_16x16x4_f32(false, aa, false, bb_l,
                                              (short)0, c, false, false);
    c = __builtin_amdgcn_wmma_f32_16x16x4_f32(false, ax, false, bb_r,
                                              (short)0, c, false, false);
  }

  // epilogue: bias + relu + store + BN partial sums
  const float bias = blv[j];
  const int   half = lane >> 4;
  float s = 0.0f, s2 = 0.0f;
#pragma unroll
  for (int r = 0; r < 8; ++r) {
    int   m = m0 + r + half * 8;
    float v = c[r] + bias;
    v = fmaxf(v, 0.0f);
    hout[(long long)m * HH + j] = v;
    s  += v;
    s2 += v * v;
  }
  atomicAdd(&bnsum[j], s);
  atomicAdd(&bnsq[j], s2);
}

// ----------------------------------------------------------- BatchNorm apply
__global__ void bn_kernel(float* __restrict__ h,
                          const float* __restrict__ bnsum,
                          const float* __restrict__ bnsq,
                          const float* __restrict__ gamma,
                          const float* __restrict__ beta) {
  long long idx = (long long)blockIdx.x * blockDim.x + threadIdx.x;
  if (idx >= (long long)NN * HH / 4) return;
  int c4 = ((int)(idx & 31)) << 2;   // feature base (HH/4 = 32 chunks per row)
  long long base = idx << 2;
  float4 v = *(float4*)(h + base);
  const float invN = 1.0f / (float)NN;
  float m0 = bnsum[c4 + 0] * invN, m1 = bnsum[c4 + 1] * invN;
  float m2 = bnsum[c4 + 2] * invN, m3 = bnsum[c4 + 3] * invN;
  float i0 = rsqrtf(bnsq[c4 + 0] * invN - m0 * m0 + BN_EPS);
  float i1 = rsqrtf(bnsq[c4 + 1] * invN - m1 * m1 + BN_EPS);
  float i2 = rsqrtf(bnsq[c4 + 2] * invN - m2 * m2 + BN_EPS);
  float i3 = rsqrtf(bnsq[c4 + 3] * invN - m3 * m3 + BN_EPS);
  v.x = (v.x - m0) * i0 * gamma[c4 + 0] + beta[c4 + 0];
  v.y = (v.y - m1) * i1 * gamma[c4 + 1] + beta[c4 + 1];
  v.z = (v.z - m2) * i2 * gamma[c4 + 2] + beta[c4 + 2];
  v.w = (v.w - m3) * i3 * gamma[c4 + 3] + beta[c4 + 3];
  *(float4*)(h + base) = v;
}

// -------------------------------------------------------------- graph pooling
__global__ void pool_kernel(const float* __restrict__ h,
                            const int* __restrict__ batch,
                            float* __restrict__ pooled) {
  long long idx = (long long)blockIdx.x * blockDim.x + threadIdx.x;
  if (idx >= (long long)NN * HH / 4) return;
  int n  = (int)(idx >> 5);
  int c4 = ((int)idx & 31) << 2;
  int g  = batch[n];
  const float4 v = *(const float4*)(h + (long long)n * HH + c4);
  float* p = pooled + (long long)g * HH + c4;
  atomicAdd(p + 0, v.x);
  atomicAdd(p + 1, v.y);
  atomicAdd(p + 2, v.z);
  atomicAdd(p + 3, v.w);
}

// ------------------------------------------------------------------ FC + sigmoid
__global__ void head_kernel(const float* __restrict__ pooled,
                            const float* __restrict__ fcw,
                            const float* __restrict__ fcb,
                            float* __restrict__ out) {
  int g = threadIdx.x;
  if (g < GG) {
    float acc = fcb[0];
#pragma unroll 4
    for (int k = 0; k < HH; ++k) acc += pooled[g * HH + k] * fcw[k];
    out[g] = 1.0f / (1.0f + expf(-acc));
  }
}

// =============================================================== entry point
extern "C" void kernel_launch(void* const* d_in, const int* in_sizes, int n_in,
                              void* d_out, int out_size, void* d_ws, size_t ws_size,
                              hipStream_t stream) {
  const float* x       = (const float*)d_in[0];
  const int*   ei      = (const int*)d_in[1];   // [2, E]
  // d_in[2] = edge_attr (unused by reference math)
  const int*   batch   = (const int*)d_in[3];
  const float* lin_l_w = (const float*)d_in[4]; // [L,H,H]
  const float* lin_l_b = (const float*)d_in[5]; // [L,H]
  const float* lin_r_w = (const float*)d_in[6]; // [L,H,H]
  const float* gamma   = (const float*)d_in[7]; // [L,H]
  const float* beta    = (const float*)d_in[8]; // [L,H]
  const float* fcw     = (const float*)d_in[9]; // [1,H]
  const float* fcb     = (const float*)d_in[10];
  float* out = (float*)d_out;

  // workspace layout (floats)
  float* hA     = (float*)d_ws;
  float* hB     = hA + (size_t)NN * HH;
  float* agg    = hB + (size_t)NN * HH;
  float* deg    = agg + (size_t)NN * HH;
  float* bnsum  = deg + NN;
  float* bnsq   = bnsum + HH;
  float* pooled = bnsq + HH;

  const int T = 256;
  // degrees (once)
  zero_f32<<<(NN + T - 1) / T, T, 0, stream>>>(deg, NN);
  deg_kernel<<<(EE + T - 1) / T, T, 0, stream>>>(ei, deg);

  const float* cur = x;
  float* nxt = hA;
  for (int l = 0; l < LL; ++l) {
    // zero agg + BN accumulators (bnsum and bnsq are contiguous: 2*HH)
    zero_f32<<<((NN * HH) + T - 1) / T, T, 0, stream>>>(agg, NN * HH);
    zero_f32<<<(2 * HH + T - 1) / T, T, 0, stream>>>(bnsum, 2 * HH);

    scatter_kernel<<<(int)(((long long)EE * 32 + T - 1) / T), T, 0, stream>>>(cur, ei, agg);

    sage_gemm_kernel<<<NN / 16, T, 0, stream>>>(
        cur, agg, deg,
        lin_l_w + (size_t)l * HH * HH, lin_l_b + (size_t)l * HH,
        lin_r_w + (size_t)l * HH * HH,
        nxt, bnsum, bnsq);

    bn_kernel<<<(int)(((long long)NN * HH / 4 + T - 1) / T), T, 0, stream>>>(
        nxt, bnsum, bnsq, gamma + (size_t)l * HH, beta + (size_t)l * HH);

    cur = nxt;
    nxt = (l == 0) ? hB : hA;
  }

  zero_f32<<<(GG * HH + T - 1) / T, T, 0, stream>>>(pooled, GG * HH);
  pool_kernel<<<(int)(((long long)NN * HH / 4 + T - 1) / T), T, 0, stream>>>(cur, batch, pooled);
  head_kernel<<<1, 64, 0, stream>>>(pooled, fcw, fcb, out);
}